// GATv2_5454608466160
// MI455X (gfx1250) — compile-verified
//
#include <hip/hip_runtime.h>

// ---------------- problem constants (match reference) ----------------
#define Nn     10000
#define Ee     160000
#define HCn    512          // H*C
#define Hh     8
#define FE     8
#define SLOPE  0.2f
#define EPSn   1e-16f

typedef float v2f __attribute__((ext_vector_type(2)));
typedef float v8f __attribute__((ext_vector_type(8)));
typedef int   v4i __attribute__((vector_size(16)));

__device__ __forceinline__ float lrelu(float x) { return x > 0.f ? x : SLOPE * x; }

// float atomic max via signed/unsigned int punning (location init'd to -inf).
__device__ __forceinline__ void atomicMaxF(float* addr, float val) {
    if (val >= 0.f) atomicMax((int*)addr, __float_as_int(val));
    else            atomicMin((unsigned int*)addr, __float_as_uint(val));
}

// ---------------- async global->LDS staging (gfx1250 path) -----------
#if __has_builtin(__builtin_amdgcn_global_load_async_to_lds_b128)
#define HAVE_ASYNC 1
#else
#define HAVE_ASYNC 0
#endif

// copy 16 contiguous bytes global -> LDS (per-lane addresses)
__device__ __forceinline__ void cp16(float* l, const float* g) {
#if HAVE_ASYNC
    __builtin_amdgcn_global_load_async_to_lds_b128(
        (__attribute__((address_space(1))) v4i*)(g),
        (__attribute__((address_space(3))) v4i*)(l), 0, 0);
#else
    *(float4*)l = *(const float4*)g;
#endif
}

__device__ __forceinline__ void async_wait_all() {
#if HAVE_ASYNC
#if __has_builtin(__builtin_amdgcn_s_wait_asynccnt)
    __builtin_amdgcn_s_wait_asynccnt(0);
#else
    asm volatile("s_wait_asynccnt 0x0" ::: "memory");
#endif
#endif
}

// ---------------------------------------------------------------------
// GEMM: Y[N x 512] = X[N x K] * W[K x 512] + bias via V_WMMA_F32_16X16X4_F32.
// Block = 128 threads (4 waves) -> 16 rows x 256 cols; each wave owns four
// 16x16 C tiles (A fragment reused 4x). K-panels of 16 staged in LDS with
// async b128 copies.
//   Xs stride 20  : A-frag ds_load_b64 conflict-free (20*r banks distinct,
//                   halves offset by 2 -> disjoint mod-4 classes).
//   Ws stride 264 : B-frag ds_load_b32, lane half K offset +2 -> +528 = +16
//                   banks, so lanes 0..31 hit 32 distinct banks.
// Fragment layouts (wave32):
//   A 16x4 f32 : lanes 0-15 K={k,k+1} (M=lane), lanes 16-31 K={k+2,k+3}.
//   B 4x16 f32 : VGPR0 rows K={k,k+2}, VGPR1 rows K={k+1,k+3}, N=lane&15.
//   C/D 16x16  : vgpr v = row m0+v (lanes 0-15) / m0+8+v (lanes 16-31).
// ---------------------------------------------------------------------
#define KP   16
#define BN   256
#define WST  264
#define XST  20

__global__ __launch_bounds__(128)
void gemm_wmma_f32(const float* __restrict__ X, const float* __restrict__ W,
                   const float* __restrict__ bias, float* __restrict__ Y, int K)
{
    __shared__ __align__(16) float Xs[16 * XST];
    __shared__ __align__(16) float Ws[KP * WST];

    const int tid    = threadIdx.x;
    const int wave   = tid >> 5;
    const int lane   = tid & 31;
    const int m0     = blockIdx.y * 16;
    const int nb     = blockIdx.x * BN;
    const int row    = lane & 15;
    const int khalf  = lane >> 4;
    const int colsub = lane & 15;

    v8f acc[4] = {};

    for (int kp = 0; kp < K; kp += KP) {
        // ---- stage X panel: 16 rows x 16 k = 64 chunks of 16B ----
        if (tid < 64) {
            const int r = tid >> 2, kc = (tid & 3) * 4;
            cp16(&Xs[r * XST + kc], &X[(m0 + r) * K + kp + kc]);
        }
        // ---- stage W panel: 16 k-rows x 256 cols = 1024 chunks of 16B ----
        for (int c = tid; c < 1024; c += 128) {
            const int k = c >> 6, nch = (c & 63) * 4;
            cp16(&Ws[k * WST + nch], &W[(kp + k) * HCn + nb + nch]);
        }
        async_wait_all();
        __syncthreads();

#pragma unroll
        for (int k2 = 0; k2 < KP; k2 += 4) {
            const int ka = k2 + 2 * khalf;
            v2f a;
            a.x = Xs[row * XST + ka];
            a.y = Xs[row * XST + ka + 1];
#pragma unroll
            for (int t = 0; t < 4; ++t) {
                const int nl = wave * 64 + t * 16 + colsub;
                v2f b;
                b.x = Ws[ka * WST + nl];
                b.y = Ws[(ka + 1) * WST + nl];
                acc[t] = __builtin_amdgcn_wmma_f32_16x16x4_f32(
                             false, a, false, b, (short)0, acc[t], false, false);
            }
        }
        __syncthreads();
    }

    const int rbase = m0 + khalf * 8;
#pragma unroll
    for (int t = 0; t < 4; ++t) {
        const int ncol = nb + wave * 64 + t * 16 + colsub;
        const float bv = bias[ncol];
#pragma unroll
        for (int v = 0; v < 8; ++v)
            Y[(rbase + v) * HCn + ncol] = acc[t][v] + bv;
    }
}

// out = 0, smax = -inf, denom = 0
__global__ void init_layer(float* __restrict__ out, float* __restrict__ smax,
                           float* __restrict__ denom)
{
    const int i = blockIdx.x * blockDim.x + threadIdx.x;
    if (i < Nn * HCn) out[i] = 0.f;
    if (i < Nn * Hh) { smax[i] = __int_as_float(0xff800000); denom[i] = 0.f; }
}

// One wave per edge: logit[e,h] = sum_c lrelu(xl[src,c]+xr[dst,c]+ee[e,c])*att[c]
// ee computed on the fly from edge_attr (8) x We (8x512 in LDS).
// Lane owns 16 contiguous channels (one head spans lanes 4h..4h+3).
__global__ __launch_bounds__(256)
void edge_logits(const int* __restrict__ ei, const float* __restrict__ eattr,
                 const float* __restrict__ XL, const float* __restrict__ XR,
                 const float* __restrict__ We, const float* __restrict__ att,
                 float* __restrict__ smax, float* __restrict__ elog)
{
    __shared__ __align__(16) float sAtt[HCn];
    __shared__ __align__(16) float sWe[FE * HCn];
    const int tid = threadIdx.x;
    for (int i = tid; i < HCn; i += 256)      sAtt[i] = att[i];
    for (int i = tid; i < FE * HCn; i += 256) sWe[i]  = We[i];
    __syncthreads();

    const int wave = tid >> 5, lane = tid & 31;
    const int e = blockIdx.x * 8 + wave;
    if (e >= Ee) return;
    const int src = ei[e], dst = ei[Ee + e];

    float ea[FE];
    const float* eap = eattr + e * FE;
#pragma unroll
    for (int j = 0; j < FE; ++j) ea[j] = eap[j];

    const int cbase = lane * 16;
    const float4* xl4 = (const float4*)(XL + (size_t)src * HCn + cbase);
    const float4* xr4 = (const float4*)(XR + (size_t)dst * HCn + cbase);

    float accv = 0.f;
#pragma unroll
    for (int q = 0; q < 4; ++q) {
        const int cg = cbase + q * 4;
        const float4 xl = xl4[q];
        const float4 xr = xr4[q];
        float4 ee = {0.f, 0.f, 0.f, 0.f};
#pragma unroll
        for (int j = 0; j < FE; ++j) {
            const float4 w = *(const float4*)&sWe[j * HCn + cg];
            ee.x += ea[j] * w.x; ee.y += ea[j] * w.y;
            ee.z += ea[j] * w.z; ee.w += ea[j] * w.w;
        }
        const float4 at = *(const float4*)&sAtt[cg];
        accv += lrelu(xl.x + xr.x + ee.x) * at.x;
        accv += lrelu(xl.y + xr.y + ee.y) * at.y;
        accv += lrelu(xl.z + xr.z + ee.z) * at.z;
        accv += lrelu(xl.w + xr.w + ee.w) * at.w;
    }
    // reduce groups of 4 lanes -> per-head logit
    accv += __shfl_xor(accv, 1);
    accv += __shfl_xor(accv, 2);
    if ((lane & 3) == 0) {
        const int h = lane >> 2;
        elog[e * Hh + h] = accv;
        atomicMaxF(&smax[dst * Hh + h], accv);
    }
}

// ex = exp(logit - smax[dst]); denom[dst] += ex  (elog overwritten in place)
__global__ void edge_exp(const int* __restrict__ ei, float* __restrict__ elog,
                         const float* __restrict__ smax, float* __restrict__ denom)
{
    const int i = blockIdx.x * blockDim.x + threadIdx.x;
    if (i >= Ee * Hh) return;
    const int e = i >> 3, h = i & 7;
    const int dst = ei[Ee + e];
    const float v = __expf(elog[i] - smax[dst * Hh + h]);
    elog[i] = v;
    unsafeAtomicAdd(&denom[dst * Hh + h], v);
}

// One wave per edge: out[dst] += alpha * xl[src]  (16 contiguous floats/lane)
__global__ __launch_bounds__(256)
void edge_agg(const int* __restrict__ ei, const float* __restrict__ XL,
              const float* __restrict__ elog, const float* __restrict__ denom,
              float* __restrict__ out)
{
    const int tid = threadIdx.x;
    const int wave = tid >> 5, lane = tid & 31;
    const int e = blockIdx.x * 8 + wave;
    if (e >= Ee) return;
    const int src = ei[e], dst = ei[Ee + e];
    const int h = lane >> 2;
    const float alpha = elog[e * Hh + h] / (denom[dst * Hh + h] + EPSn);
    const int cbase = lane * 16;
    const float4* xl4 = (const float4*)(XL + (size_t)src * HCn + cbase);
    float* op = out + (size_t)dst * HCn + cbase;
#pragma unroll
    for (int q = 0; q < 4; ++q) {
        const float4 xl = xl4[q];
        unsafeAtomicAdd(&op[q * 4 + 0], alpha * xl.x);
        unsafeAtomicAdd(&op[q * 4 + 1], alpha * xl.y);
        unsafeAtomicAdd(&op[q * 4 + 2], alpha * xl.z);
        unsafeAtomicAdd(&op[q * 4 + 3], alpha * xl.w);
    }
}

// h = lrelu(h + bo)
__global__ void epilogue(float* __restrict__ h, const float* __restrict__ bo)
{
    const int i = blockIdx.x * blockDim.x + threadIdx.x;
    if (i >= Nn * HCn) return;
    h[i] = lrelu(h[i] + bo[i & (HCn - 1)]);
}

// out[n, 0..3] = h[n,:] @ Wf + bf ; one wave per node
__global__ __launch_bounds__(256)
void classifier(const float* __restrict__ h, const float* __restrict__ Wf,
                const float* __restrict__ bf, float* __restrict__ out)
{
    const int tid = threadIdx.x;
    const int wave = tid >> 5, lane = tid & 31;
    const int n = blockIdx.x * 8 + wave;
    if (n >= Nn) return;
    float p0 = 0.f, p1 = 0.f, p2 = 0.f, p3 = 0.f;
    const float* hp = h + (size_t)n * HCn;
    for (int k = lane; k < HCn; k += 32) {
        const float xv = hp[k];
        const float4 w = *(const float4*)&Wf[k * 4];
        p0 += xv * w.x; p1 += xv * w.y; p2 += xv * w.z; p3 += xv * w.w;
    }
#pragma unroll
    for (int off = 16; off; off >>= 1) {
        p0 += __shfl_xor(p0, off);
        p1 += __shfl_xor(p1, off);
        p2 += __shfl_xor(p2, off);
        p3 += __shfl_xor(p3, off);
    }
    if (lane == 0) {
        out[n * 4 + 0] = p0 + bf[0];
        out[n * 4 + 1] = p1 + bf[1];
        out[n * 4 + 2] = p2 + bf[2];
        out[n * 4 + 3] = p3 + bf[3];
    }
}

// ---------------------------------------------------------------------
extern "C" void kernel_launch(void* const* d_in, const int* in_sizes, int n_in,
                              void* d_out, int out_size, void* d_ws, size_t ws_size,
                              hipStream_t stream)
{
    (void)in_sizes; (void)n_in; (void)out_size; (void)ws_size;

    const float* x  = (const float*)d_in[0];
    const int*   ei = (const int*)  d_in[1];
    const float* ea = (const float*)d_in[2];
    const float* Wf = (const float*)d_in[24];
    const float* bf = (const float*)d_in[25];

    float* ws    = (float*)d_ws;
    float* outA  = ws;                       // N*512
    float* outB  = ws + (size_t)1 * Nn * HCn;
    float* XLb   = ws + (size_t)2 * Nn * HCn;
    float* XRb   = ws + (size_t)3 * Nn * HCn;
    float* smax  = ws + (size_t)4 * Nn * HCn;
    float* denom = smax  + (size_t)Nn * Hh;
    float* elog  = denom + (size_t)Nn * Hh;

    const dim3 gemmGrid(HCn / BN, Nn / 16);           // (2, 625)
    const int  nhcBlocks  = (Nn * HCn + 255) / 256;   // 20000
    const int  edgeBlocks = (Ee + 7) / 8;             // 20000 (8 waves/block)
    const int  ehBlocks   = (Ee * Hh + 255) / 256;    // 5000

    for (int l = 0; l < 3; ++l) {
        const float* Wl  = (const float*)d_in[3 + 7 * l + 0];
        const float* bl  = (const float*)d_in[3 + 7 * l + 1];
        const float* Wr  = (const float*)d_in[3 + 7 * l + 2];
        const float* br  = (const float*)d_in[3 + 7 * l + 3];
        const float* Wel = (const float*)d_in[3 + 7 * l + 4];
        const float* atl = (const float*)d_in[3 + 7 * l + 5];
        const float* bo  = (const float*)d_in[3 + 7 * l + 6];

        const float* hin = (l == 0) ? x : (l == 1 ? outA : outB);
        float*       out = (l == 1) ? outB : outA;   // l0->A, l1->B, l2->A
        const int    K   = (l == 0) ? 16 : HCn;

        gemm_wmma_f32<<<gemmGrid, 128, 0, stream>>>(hin, Wl, bl, XLb, K);
        gemm_wmma_f32<<<gemmGrid, 128, 0, stream>>>(hin, Wr, br, XRb, K);
        init_layer  <<<nhcBlocks, 256, 0, stream>>>(out, smax, denom);
        edge_logits <<<edgeBlocks, 256, 0, stream>>>(ei, ea, XLb, XRb, Wel, atl, smax, elog);
        edge_exp    <<<ehBlocks, 256, 0, stream>>>(ei, elog, smax, denom);
        edge_agg    <<<edgeBlocks, 256, 0, stream>>>(ei, XLb, elog, denom, out);
        epilogue    <<<nhcBlocks, 256, 0, stream>>>(out, bo);
    }

    classifier<<<(Nn + 7) / 8, 256, 0, stream>>>(outA, Wf, bf, (float*)d_out);
}